// PathConvStack_54022098649979
// MI455X (gfx1250) — compile-verified
//
#include <hip/hip_runtime.h>
#include <math.h>

// ---------------------------------------------------------------------------
// Problem constants (match reference)
// ---------------------------------------------------------------------------
#define PAIR_CHANNELS 128
#define N_HEADS 8
#define MAX_PI_LENGTH 8
#define NUM_IJKL 1000000
#define NUM_IJK 200000
#define NUM_UIJK 400000
#define NUM_UIJKL 3000000
#define DCAT 72          // N_HEADS * (MAX_PI_LENGTH + 1)

typedef __attribute__((ext_vector_type(2))) float v2f;
typedef __attribute__((ext_vector_type(8))) float v8f;

// f32 WMMA: D = A(16x4) * B(4x16) + C(16x16), full fp32 precision
__device__ inline v8f wmma_f32_4(v2f a, v2f b, v8f c) {
    return __builtin_amdgcn_wmma_f32_16x16x4_f32(
        /*neg_a=*/false, a, /*neg_b=*/false, b,
        /*c_mod=*/(short)0, c, /*reuse_a=*/false, /*reuse_b=*/false);
}

// float atomic-max via monotonic int encoding (positive: int max, negative: uint min)
__device__ inline void atomicMaxF(float* addr, float val) {
    if (val >= 0.0f) {
        atomicMax((int*)addr, __float_as_int(val));
    } else {
        atomicMin((unsigned int*)addr, (unsigned int)__float_as_int(val));
    }
}

// ---------------------------------------------------------------------------
// 0) Workspace init: x = 0, segsum = 0, segmax = -inf
// ---------------------------------------------------------------------------
__global__ __launch_bounds__(256) void init_ws_kernel(float* __restrict__ segmax,
                                                      float* __restrict__ segsum,
                                                      float* __restrict__ x) {
    const int nseg = NUM_IJK * N_HEADS;        // 1.6M
    const int nx   = NUM_UIJK * DCAT;          // 28.8M
    int i = blockIdx.x * blockDim.x + threadIdx.x;
    if (i < nx) x[i] = 0.0f;
    if (i < nseg) {
        segmax[i] = -INFINITY;
        segsum[i] = 0.0f;
    }
}

// ---------------------------------------------------------------------------
// 1) Tall-skinny GEMM via f32 WMMA: out[r, 0..7] = A[r, 0..127] @ W[128, 8]
//    One wave per 16-row tile, 4 waves per block. N-tile of 16 with cols 8..15
//    masked to zero (clamped index * mask keeps EXEC uniform for WMMA).
// ---------------------------------------------------------------------------
__global__ __launch_bounds__(128) void gemm_skinny_kernel(const float* __restrict__ A,
                                                          const float* __restrict__ W,
                                                          float* __restrict__ out,
                                                          int out_stride, int nrows) {
    const int wave = threadIdx.x >> 5;
    const int lane = threadIdx.x & 31;
    const int m  = lane & 15;      // A-frag row / D col index
    const int hi = lane >> 4;      // half-wave selector

    const int row0 = (blockIdx.x * 4 + wave) * 16;
    if (row0 >= nrows) return;     // wave-uniform

    const int  n     = m;                       // D/B column within n-tile
    const int  nc    = (n < N_HEADS) ? n : 0;   // clamped col
    const float mask = (n < N_HEADS) ? 1.0f : 0.0f;

    v8f acc = {};
    const float2* arow = (const float2*)(A + (size_t)(row0 + m) * PAIR_CHANNELS);
#pragma unroll
    for (int kc = 0; kc < PAIR_CHANNELS / 4; ++kc) {   // 32 k-chunks of 4
        const int k0 = kc * 4 + 2 * hi;                // even
        float2 av = arow[k0 >> 1];
        v2f a; a.x = av.x; a.y = av.y;
        v2f b;
        b.x = W[k0 * N_HEADS + nc] * mask;
        b.y = W[(k0 + 1) * N_HEADS + nc] * mask;
        acc = wmma_f32_4(a, b, acc);
    }

    if (n < N_HEADS) {
#pragma unroll
        for (int r = 0; r < 8; ++r) {
            int row = row0 + r + 8 * hi;
            out[(size_t)row * out_stride + n] = acc[r];
        }
    }
}

// ---------------------------------------------------------------------------
// 2) Segment softmax (3 passes over [NUM_IJKL, 8])
// ---------------------------------------------------------------------------
__global__ __launch_bounds__(256) void seg_max_kernel(const float* __restrict__ logits,
                                                      const int* __restrict__ seg,
                                                      float* __restrict__ segmax) {
    int u = blockIdx.x * blockDim.x + threadIdx.x;
    if (u >= NUM_IJKL) return;
    int s = seg[u];
    const float4* lp = (const float4*)(logits + (size_t)u * N_HEADS);
    float4 l0 = lp[0], l1 = lp[1];
    float* mrow = segmax + (size_t)s * N_HEADS;
    atomicMaxF(mrow + 0, l0.x); atomicMaxF(mrow + 1, l0.y);
    atomicMaxF(mrow + 2, l0.z); atomicMaxF(mrow + 3, l0.w);
    atomicMaxF(mrow + 4, l1.x); atomicMaxF(mrow + 5, l1.y);
    atomicMaxF(mrow + 6, l1.z); atomicMaxF(mrow + 7, l1.w);
}

__global__ __launch_bounds__(256) void seg_expsum_kernel(float* __restrict__ logits,
                                                         const int* __restrict__ seg,
                                                         const float* __restrict__ segmax,
                                                         float* __restrict__ segsum) {
    int u = blockIdx.x * blockDim.x + threadIdx.x;
    if (u >= NUM_IJKL) return;
    int s = seg[u];
    float4* lp = (float4*)(logits + (size_t)u * N_HEADS);
    const float4* mp = (const float4*)(segmax + (size_t)s * N_HEADS);
    float4 l0 = lp[0], l1 = lp[1];
    float4 m0 = mp[0], m1 = mp[1];
    float4 e0, e1;
    e0.x = expf(l0.x - m0.x); e0.y = expf(l0.y - m0.y);
    e0.z = expf(l0.z - m0.z); e0.w = expf(l0.w - m0.w);
    e1.x = expf(l1.x - m1.x); e1.y = expf(l1.y - m1.y);
    e1.z = expf(l1.z - m1.z); e1.w = expf(l1.w - m1.w);
    lp[0] = e0; lp[1] = e1;  // logits buffer now holds e
    float* srow = segsum + (size_t)s * N_HEADS;
    atomicAdd(srow + 0, e0.x); atomicAdd(srow + 1, e0.y);
    atomicAdd(srow + 2, e0.z); atomicAdd(srow + 3, e0.w);
    atomicAdd(srow + 4, e1.x); atomicAdd(srow + 5, e1.y);
    atomicAdd(srow + 6, e1.z); atomicAdd(srow + 7, e1.w);
}

__global__ __launch_bounds__(256) void seg_norm_kernel(float* __restrict__ e,
                                                       const int* __restrict__ seg,
                                                       const float* __restrict__ segsum) {
    int u = blockIdx.x * blockDim.x + threadIdx.x;
    if (u >= NUM_IJKL) return;
    int s = seg[u];
    float4* ep = (float4*)(e + (size_t)u * N_HEADS);
    const float4* sp = (const float4*)(segsum + (size_t)s * N_HEADS);
    float4 e0 = ep[0], e1 = ep[1];
    float4 s0 = sp[0], s1 = sp[1];
    e0.x /= s0.x; e0.y /= s0.y; e0.z /= s0.z; e0.w /= s0.w;
    e1.x /= s1.x; e1.y /= s1.y; e1.z /= s1.z; e1.w /= s1.w;
    ep[0] = e0; ep[1] = e1;   // e buffer now holds alpha
}

// 3) kernelv[u, :] = alpha[gather_idx_Uijkl_ijkl[u], :]
__global__ __launch_bounds__(256) void gather_kv_kernel(const float* __restrict__ alpha,
                                                        const int* __restrict__ gidx,
                                                        float* __restrict__ kernelv) {
    int u = blockIdx.x * blockDim.x + threadIdx.x;
    if (u >= NUM_UIJKL) return;
    int j = gidx[u];
    const float4* ap = (const float4*)(alpha + (size_t)j * N_HEADS);
    float4* kp = (float4*)(kernelv + (size_t)u * N_HEADS);
    kp[0] = ap[0];
    kp[1] = ap[1];
}

// ---------------------------------------------------------------------------
// 4) Propagation round t: x[:, (t+1)*8 : (t+2)*8] += segment_sum(kernel * g)
//    float4 gathers (rows are 32B-aligned: 72 = 8*9), scalar f32 atomics out.
// ---------------------------------------------------------------------------
__global__ __launch_bounds__(256) void propagate_kernel(const float* __restrict__ kernelv,
                                                        float* __restrict__ x,
                                                        const int* __restrict__ g_src,
                                                        const int* __restrict__ g_dst,
                                                        int col_in, int col_out) {
    int u = blockIdx.x * blockDim.x + threadIdx.x;
    if (u >= NUM_UIJKL) return;
    const float4* kp = (const float4*)(kernelv + (size_t)u * N_HEADS);
    float4 k0 = kp[0], k1 = kp[1];
    int src = g_src[u];
    int dst = g_dst[u];
    const float4* gp = (const float4*)(x + (size_t)src * DCAT + col_in);
    float4 g0 = gp[0], g1 = gp[1];
    float* o = x + (size_t)dst * DCAT + col_out;
    atomicAdd(o + 0, k0.x * g0.x); atomicAdd(o + 1, k0.y * g0.y);
    atomicAdd(o + 2, k0.z * g0.z); atomicAdd(o + 3, k0.w * g0.w);
    atomicAdd(o + 4, k1.x * g1.x); atomicAdd(o + 5, k1.y * g1.y);
    atomicAdd(o + 6, k1.z * g1.z); atomicAdd(o + 7, k1.w * g1.w);
}

// ---------------------------------------------------------------------------
// 5) Fused LayerNorm -> GEMM1(72x72) -> exact GELU -> GEMM2(72x128) -> +prop
//    One wave per 16-row tile; 4 waves (128 threads) per block; WMMA f32.
// ---------------------------------------------------------------------------
__global__ __launch_bounds__(128) void mlp_kernel(const float* __restrict__ x,
                                                  const float* __restrict__ prop,
                                                  const float* __restrict__ W1,
                                                  const float* __restrict__ b1,
                                                  const float* __restrict__ W2,
                                                  const float* __restrict__ b2,
                                                  const float* __restrict__ ln_scale,
                                                  const float* __restrict__ ln_bias,
                                                  float* __restrict__ out) {
    __shared__ float sx[4][16][80];   // LN'd activations, padded stride
    __shared__ float sh[4][16][80];   // GELU hidden

    const int wave = threadIdx.x >> 5;
    const int lane = threadIdx.x & 31;
    const int m  = lane & 15;
    const int hi = lane >> 4;
    const int row0 = (blockIdx.x * 4 + wave) * 16;

    // ---- cooperative load of 16x72 tile ----
    for (int idx = lane; idx < 16 * DCAT; idx += 32) {
        int r = idx / DCAT, c = idx % DCAT;
        sx[wave][r][c] = x[(size_t)(row0 + r) * DCAT + c];
    }
    __syncthreads();

    // ---- LayerNorm: 2 lanes per row, 36 elements each ----
    {
        int row  = lane >> 1;
        int half = lane & 1;
        float s = 0.0f, sq = 0.0f;
#pragma unroll
        for (int j = 0; j < 36; ++j) {
            float v = sx[wave][row][half * 36 + j];
            s += v; sq += v * v;
        }
        s  += __shfl_xor(s, 1);
        sq += __shfl_xor(sq, 1);
        float mean = s * (1.0f / DCAT);
        float var  = sq * (1.0f / DCAT) - mean * mean;
        float inv  = rsqrtf(var + 1e-5f);
#pragma unroll
        for (int j = 0; j < 36; ++j) {
            int c = half * 36 + j;
            float v = sx[wave][row][c];
            sx[wave][row][c] = (v - mean) * inv * ln_scale[c] + ln_bias[c];
        }
    }
    __syncthreads();

    const int n = m;   // D/B column within an n-tile

    // ---- GEMM1: h = GELU( LN(x) @ W1 + b1 ), N=72 padded to 5 n-tiles ----
#pragma unroll
    for (int nt = 0; nt < 5; ++nt) {
        int col = nt * 16 + n;
        int cc = (col < DCAT) ? col : 0;
        float msk = (col < DCAT) ? 1.0f : 0.0f;
        v8f acc = {};
#pragma unroll
        for (int kc = 0; kc < DCAT / 4; ++kc) {        // 18 k-chunks
            int k0 = kc * 4 + 2 * hi;
            v2f a; a.x = sx[wave][m][k0]; a.y = sx[wave][m][k0 + 1];
            v2f b;
            b.x = W1[k0 * DCAT + cc] * msk;
            b.y = W1[(k0 + 1) * DCAT + cc] * msk;
            acc = wmma_f32_4(a, b, acc);
        }
        if (col < DCAT) {
            float bb = b1[col];
#pragma unroll
            for (int r = 0; r < 8; ++r) {
                float v = acc[r] + bb;
                v = 0.5f * v * (1.0f + erff(v * 0.70710678118654752f));  // exact GELU
                sh[wave][r + 8 * hi][col] = v;
            }
        }
    }
    __syncthreads();

    // ---- GEMM2: y = h @ W2 + b2 + prop, N=128 -> 8 n-tiles ----
#pragma unroll
    for (int nt = 0; nt < 8; ++nt) {
        int col = nt * 16 + n;
        v8f acc = {};
#pragma unroll
        for (int kc = 0; kc < DCAT / 4; ++kc) {
            int k0 = kc * 4 + 2 * hi;
            v2f a; a.x = sh[wave][m][k0]; a.y = sh[wave][m][k0 + 1];
            v2f b;
            b.x = W2[k0 * PAIR_CHANNELS + col];
            b.y = W2[(k0 + 1) * PAIR_CHANNELS + col];
            acc = wmma_f32_4(a, b, acc);
        }
        float bb = b2[col];
#pragma unroll
        for (int r = 0; r < 8; ++r) {
            int row = row0 + r + 8 * hi;
            size_t o = (size_t)row * PAIR_CHANNELS + col;
            out[o] = acc[r] + bb + prop[o];
        }
    }
}

// ---------------------------------------------------------------------------
// Host orchestration
// ---------------------------------------------------------------------------
extern "C" void kernel_launch(void* const* d_in, const int* in_sizes, int n_in,
                              void* d_out, int out_size, void* d_ws, size_t ws_size,
                              hipStream_t stream) {
    const float* prop     = (const float*)d_in[0];
    const float* stereo   = (const float*)d_in[1];
    const int*   g_jkl    = (const int*)d_in[2];
    const int*   g_ijkl   = (const int*)d_in[3];
    const int*   g_Uijk   = (const int*)d_in[4];
    const int*   g_ujkl   = (const int*)d_in[5];
    // d_in[6], d_in[7] are num_ijk / num_Uijk scalars (compile-time constants here)
    const float* W_value  = (const float*)d_in[8];
    const float* W_kernel = (const float*)d_in[9];
    const float* ln_scale = (const float*)d_in[10];
    const float* ln_bias  = (const float*)d_in[11];
    const float* W1       = (const float*)d_in[12];
    const float* b1       = (const float*)d_in[13];
    const float* W2       = (const float*)d_in[14];
    const float* b2       = (const float*)d_in[15];
    float* out = (float*)d_out;

    // workspace carve (≈256 MB total)
    char* p = (char*)d_ws;
    float* logits  = (float*)p; p += (size_t)NUM_IJKL * N_HEADS * sizeof(float);   // 32 MB
    float* segmax  = (float*)p; p += (size_t)NUM_IJK  * N_HEADS * sizeof(float);   // 6.4 MB
    float* segsum  = (float*)p; p += (size_t)NUM_IJK  * N_HEADS * sizeof(float);   // 6.4 MB
    float* kernelv = (float*)p; p += (size_t)NUM_UIJKL * N_HEADS * sizeof(float);  // 96 MB
    float* x       = (float*)p;                                                    // 115.2 MB

    const int nx = NUM_UIJK * DCAT;

    // 0) init
    init_ws_kernel<<<(nx + 255) / 256, 256, 0, stream>>>(segmax, segsum, x);

    // 1) kernel logits = stereo @ W_kernel  (WMMA)
    gemm_skinny_kernel<<<NUM_IJKL / 64, 128, 0, stream>>>(stereo, W_kernel, logits,
                                                          N_HEADS, NUM_IJKL);
    // 2) segment softmax
    seg_max_kernel<<<(NUM_IJKL + 255) / 256, 256, 0, stream>>>(logits, g_jkl, segmax);
    seg_expsum_kernel<<<(NUM_IJKL + 255) / 256, 256, 0, stream>>>(logits, g_jkl, segmax, segsum);
    seg_norm_kernel<<<(NUM_IJKL + 255) / 256, 256, 0, stream>>>(logits, g_jkl, segsum);

    // 3) gather kernel per Uijkl edge
    gather_kv_kernel<<<(NUM_UIJKL + 255) / 256, 256, 0, stream>>>(logits, g_ijkl, kernelv);

    // 4) outs[0] = prop @ W_value, written into x columns 0..7 (WMMA)
    gemm_skinny_kernel<<<NUM_UIJK / 64, 128, 0, stream>>>(prop, W_value, x, DCAT, NUM_UIJK);

    // 5) 8 propagation rounds: x[:, (t+1)*8..] = segsum(kernel * gather(x[:, t*8..]))
    for (int t = 0; t < MAX_PI_LENGTH; ++t) {
        propagate_kernel<<<(NUM_UIJKL + 255) / 256, 256, 0, stream>>>(
            kernelv, x, g_Uijk, g_ujkl, t * N_HEADS, (t + 1) * N_HEADS);
    }

    // 6) LN -> MLP -> residual (WMMA)
    mlp_kernel<<<NUM_UIJK / 64, 128, 0, stream>>>(x, prop, W1, b1, W2, b2,
                                                  ln_scale, ln_bias, out);
}